// EdgeConv_41394894798866
// MI455X (gfx1250) — compile-verified
//
#include <hip/hip_runtime.h>
#include <math.h>

#define BATCH 8
#define NPTS  4096
#define KNN   20
#define HCH   64
#define TOTAL (BATCH*NPTS)
#define NEDGE (TOTAL*KNN)
#define NEG_SLOPE 0.2f
#define BN_EPS    1e-5f

typedef __attribute__((ext_vector_type(2))) float v2f;
typedef __attribute__((ext_vector_type(8))) float v8f;

// ---------------------------------------------------------------------------
// Kernel 1: kNN (thread-per-query, LDS candidate tiles, register top-20)
// ---------------------------------------------------------------------------
__global__ void __launch_bounds__(128)
knn_kernel(const float* __restrict__ pcd, int* __restrict__ idxws) {
  __shared__ float spx[128], spy[128], spz[128], ssq[128];
  const int b = blockIdx.x >> 5;                 // 32 query tiles of 128 per cloud
  const int q = ((blockIdx.x & 31) << 7) + threadIdx.x;
  const float* pb = pcd + (size_t)b * NPTS * 3;
  const float qx = pb[q*3+0], qy = pb[q*3+1], qz = pb[q*3+2];
  const float qsq = qx*qx + qy*qy + qz*qz;

  float dist[KNN];
  int   nidx[KNN];
  #pragma unroll
  for (int j = 0; j < KNN; ++j) { dist[j] = 3.4e38f; nidx[j] = 0; }

  for (int t = 0; t < NPTS/128; ++t) {
    const int c = t*128 + threadIdx.x;
    const float x = pb[c*3+0], y = pb[c*3+1], z = pb[c*3+2];
    spx[threadIdx.x] = x; spy[threadIdx.x] = y; spz[threadIdx.x] = z;
    ssq[threadIdx.x] = x*x + y*y + z*z;
    __syncthreads();
    #pragma unroll 4
    for (int u = 0; u < 128; ++u) {
      const float d2 = qsq + ssq[u] - 2.f*(qx*spx[u] + qy*spy[u] + qz*spz[u]);
      if (d2 < dist[KNN-1]) {                    // insert + one bubble pass
        dist[KNN-1] = d2; nidx[KNN-1] = t*128 + u;
        #pragma unroll
        for (int j = KNN-1; j > 0; --j) {
          const float dj = dist[j], dj1 = dist[j-1];
          const int   ij = nidx[j], ij1 = nidx[j-1];
          const bool  sw = dj < dj1;
          dist[j]   = sw ? dj1 : dj;  dist[j-1] = sw ? dj  : dj1;
          nidx[j]   = sw ? ij1 : ij;  nidx[j-1] = sw ? ij  : ij1;
        }
      }
    }
    __syncthreads();
  }
  int* op = idxws + (size_t)(b*NPTS + q) * KNN;
  #pragma unroll
  for (int j = 0; j < KNN; ++j) op[j] = nidx[j];
}

// ---------------------------------------------------------------------------
// Kernel 2: feature moments — Sum(f) [6] and Sum(f f^T) upper triangle [21].
// BN mean/var per channel derive from these (h = f.W + b is affine).
// ---------------------------------------------------------------------------
__global__ void __launch_bounds__(256)
moments_kernel(const float* __restrict__ pcd, const int* __restrict__ idxws,
               float* __restrict__ stats) {
  __shared__ float sacc[27];
  if (threadIdx.x < 27) sacc[threadIdx.x] = 0.f;
  __syncthreads();

  float s[6], m[21];
  #pragma unroll
  for (int j = 0; j < 6;  ++j) s[j] = 0.f;
  #pragma unroll
  for (int j = 0; j < 21; ++j) m[j] = 0.f;

  for (int pt = blockIdx.x*blockDim.x + threadIdx.x; pt < TOTAL;
       pt += gridDim.x*blockDim.x) {
    const int b = pt >> 12;
    const float sx = pcd[pt*3+0], sy = pcd[pt*3+1], sz = pcd[pt*3+2];
    const int* ip = idxws + (size_t)pt * KNN;
    for (int k = 0; k < KNN; ++k) {
      const int nj = (b << 12) + ip[k];
      const float nx = pcd[nj*3+0], ny = pcd[nj*3+1], nz = pcd[nj*3+2];
      const float f[6] = { nx, ny, nz, sx-nx, sy-ny, sz-nz };
      int c = 0;
      #pragma unroll
      for (int j = 0; j < 6; ++j) {
        s[j] += f[j];
        #pragma unroll
        for (int k2 = j; k2 < 6; ++k2) { m[c] += f[j]*f[k2]; ++c; }
      }
    }
  }
  #pragma unroll
  for (int j = 0; j < 6;  ++j) atomicAdd(&sacc[j],   s[j]);
  #pragma unroll
  for (int j = 0; j < 21; ++j) atomicAdd(&sacc[6+j], m[j]);
  __syncthreads();
  if (threadIdx.x < 27) atomicAdd(&stats[threadIdx.x], sacc[threadIdx.x]);
}

// ---------------------------------------------------------------------------
// Kernel 3: fused gather + [80x8]@[8x64] fp32 WMMA + BN affine + LeakyReLU
//           + max over K=20 edges. One wave handles 4 points (80 edges = 5
//           M-tiles of 16); block of 128 = 4 waves = 16 points.
// ---------------------------------------------------------------------------
__global__ void __launch_bounds__(128)
edgeconv_kernel(const float* __restrict__ pcd, const int* __restrict__ idxws,
                const float* __restrict__ stats, const float* __restrict__ W,
                const float* __restrict__ bias, const float* __restrict__ gamma,
                const float* __restrict__ beta, float* __restrict__ out) {
  __shared__ float sh_scale[HCH], sh_shift[HCH];
  __shared__ __align__(16) float sfeat[4*80*8];   // per-wave [80][8] fp32 (K padded)
  __shared__ float shbuf[4*80*16];                // per-wave [80][16] post-BN slab

  const int tid  = threadIdx.x;
  const int wave = tid >> 5, lane = tid & 31;

  // --- derive BN scale/shift per channel from the 27 feature moments ---
  if (tid < HCH) {
    const float invE = 1.f / (float)NEDGE;
    float wc[6];
    #pragma unroll
    for (int k = 0; k < 6; ++k) wc[k] = W[k*HCH + tid];
    float sdot = 0.f;
    #pragma unroll
    for (int k = 0; k < 6; ++k) sdot += stats[k]*wc[k];
    float quad = 0.f;
    int c = 0;
    #pragma unroll
    for (int j = 0; j < 6; ++j)
      #pragma unroll
      for (int k2 = j; k2 < 6; ++k2) {
        const float t = stats[6+c]*wc[j]*wc[k2];
        quad += (k2 == j) ? t : 2.f*t;
        ++c;
      }
    const float bc   = bias[tid];
    const float mean = sdot*invE + bc;
    const float ex2  = quad*invE + 2.f*bc*sdot*invE + bc*bc;
    const float var  = ex2 - mean*mean;
    const float scl  = gamma[tid] * rsqrtf(var + BN_EPS);
    sh_scale[tid] = scl;
    sh_shift[tid] = beta[tid] - mean*scl;
  }

  // --- stage edge features for this wave's 4 points: feat = [nbr, xi - nbr, 0, 0]
  const int pt_base = blockIdx.x*16 + wave*4;
  float* featw = sfeat + wave*80*8;
  for (int e = lane; e < 80; e += 32) {
    const int p  = e/20, kk = e - p*20;
    const int pt = pt_base + p;
    const int b  = pt >> 12;
    const int nj = (b << 12) + idxws[(size_t)pt*KNN + kk];
    const float sx = pcd[pt*3+0], sy = pcd[pt*3+1], sz = pcd[pt*3+2];
    const float nx = pcd[nj*3+0], ny = pcd[nj*3+1], nz = pcd[nj*3+2];
    float* fr = featw + e*8;
    fr[0]=nx; fr[1]=ny; fr[2]=nz; fr[3]=sx-nx; fr[4]=sy-ny; fr[5]=sz-nz;
    fr[6]=0.f; fr[7]=0.f;
  }

  // --- B fragments: W [6][64] zero-padded to K=8. ISA 16x4 f32 layout:
  //     lanes 0-15 hold K={kb,kb+1}, lanes 16-31 hold K={kb+2,kb+3}.
  const int ch    = lane & 15;
  const int khalf = lane >> 4;
  v2f bf[4][2];
  #pragma unroll
  for (int nt = 0; nt < 4; ++nt) {
    const int n = nt*16 + ch;
    #pragma unroll
    for (int ks = 0; ks < 2; ++ks) {
      const int k0 = ks*4 + khalf*2;
      v2f v;
      v.x = (k0   < 6) ? W[k0*HCH + n]     : 0.f;
      v.y = (k0+1 < 6) ? W[(k0+1)*HCH + n] : 0.f;
      bf[nt][ks] = v;
    }
  }
  __syncthreads();   // scale/shift + features visible; EXEC all-1 hereafter

  float* hbw = shbuf + wave*80*16;
  #pragma unroll
  for (int nt = 0; nt < 4; ++nt) {
    const float scl = sh_scale[nt*16 + ch];
    const float shf = sh_shift[nt*16 + ch];
    #pragma unroll
    for (int mt = 0; mt < 5; ++mt) {
      const int m = mt*16 + ch;                         // A row for this lane
      const v2f a0 = *(const v2f*)(featw + m*8 + 0 + khalf*2);
      const v2f a1 = *(const v2f*)(featw + m*8 + 4 + khalf*2);
      v8f cacc = {};
      cacc = __builtin_amdgcn_wmma_f32_16x16x4_f32(
          false, a0, false, bf[nt][0], (short)0, cacc, false, false);
      cacc = __builtin_amdgcn_wmma_f32_16x16x4_f32(
          false, a1, false, bf[nt][1], (short)0, cacc, false, false);
      #pragma unroll
      for (int j = 0; j < 8; ++j) {                     // C/D layout: M = j + 8*khalf
        const int mrow = mt*16 + j + 8*khalf;
        float v = cacc[j]*scl + shf;
        v = (v >= 0.f) ? v : NEG_SLOPE*v;
        hbw[mrow*16 + ch] = v;
      }
    }
    asm volatile("s_wait_dscnt 0" ::: "memory");        // slab stores -> pool loads
    #pragma unroll
    for (int r = 0; r < 2; ++r) {                       // 4 pts x 16 ch = 64 maxes
      const int id = lane + 32*r;
      const int p = id >> 4, c2 = id & 15;
      float mx = -3.4e38f;
      #pragma unroll
      for (int e = 0; e < KNN; ++e)
        mx = fmaxf(mx, hbw[(p*20 + e)*16 + c2]);
      out[(size_t)(pt_base + p)*HCH + nt*16 + c2] = mx;
    }
    asm volatile("s_wait_dscnt 0" ::: "memory");        // pool loads -> next slab
  }
}

// ---------------------------------------------------------------------------
extern "C" void kernel_launch(void* const* d_in, const int* in_sizes, int n_in,
                              void* d_out, int out_size, void* d_ws, size_t ws_size,
                              hipStream_t stream) {
  const float* pcd = (const float*)d_in[0];
  const float* W   = (const float*)d_in[1];
  const float* b   = (const float*)d_in[2];
  const float* gm  = (const float*)d_in[3];
  const float* bt  = (const float*)d_in[4];

  int*   idxws = (int*)d_ws;                                    // [TOTAL][20]
  float* stats = (float*)((char*)d_ws + (size_t)NEDGE*sizeof(int)); // [27]

  hipMemsetAsync(stats, 0, 27*sizeof(float), stream);           // graph-capturable
  knn_kernel     <<<BATCH*(NPTS/128), 128, 0, stream>>>(pcd, idxws);
  moments_kernel <<<64, 256, 0, stream>>>(pcd, idxws, stats);
  edgeconv_kernel<<<TOTAL/16, 128, 0, stream>>>(pcd, idxws, stats, W, b, gm, bt,
                                                (float*)d_out);
}